// Pair_Generate_68006512165078
// MI455X (gfx1250) — compile-verified
//
#include <hip/hip_runtime.h>

// ---- problem constants (fixed by reference setup_inputs) ----
#define TAGSET 7
#define B_   16
#define L_   128
#define F_   768
#define K_   12
#define PD   50          // pos_dim
#define ED   50          // emo_dim
#define NC   3044        // number of banded (i,j) pairs for L=128, K=12
#define ROW  1636        // 768 + 768 + 50 + 50
#define TPAD 64          // padded stride of pos table (32 x 64 f32)

typedef float v2f __attribute__((ext_vector_type(2)));
typedef float v8f __attribute__((ext_vector_type(8)));
typedef float f4  __attribute__((ext_vector_type(4)));

// ------------------------------------------------------------------
// Kernel 1: build band pair index lists + emo_cau_pos output
// ------------------------------------------------------------------
__global__ void build_pairs_kernel(int* __restrict__ ei, int* __restrict__ ci,
                                   float* __restrict__ out_pos)
{
    const int i = threadIdx.x;
    if (i >= L_) return;
    int off = 0;
    for (int t = 0; t < i; ++t) {
        int lo = (t - K_ < 0) ? 0 : t - K_;
        int hi = (t + K_ > L_ - 1) ? L_ - 1 : t + K_;
        off += hi - lo + 1;
    }
    const int jlo = (i - K_ < 0) ? 0 : i - K_;
    const int jhi = (i + K_ > L_ - 1) ? L_ - 1 : i + K_;
    for (int j = jlo; j <= jhi; ++j) {
        const int p = off + (j - jlo);
        ei[p] = i;
        ci[p] = j;
        out_pos[2 * p + 0] = (float)(i + 1);
        out_pos[2 * p + 1] = (float)(j + 1);
    }
}

// ------------------------------------------------------------------
// Kernel 2: argmax over TAGSET logits -> labels (first max wins)
// ------------------------------------------------------------------
__global__ void labels_kernel(const float* __restrict__ pred,
                              int* __restrict__ labels, int n)
{
    const int idx = blockIdx.x * blockDim.x + threadIdx.x;
    if (idx >= n) return;
    const float* r = pred + (size_t)idx * TAGSET;
    float best = r[0];
    int bi = 0;
    #pragma unroll
    for (int k = 1; k < TAGSET; ++k) {
        float v = r[k];
        if (v > best) { best = v; bi = k; }
    }
    labels[idx] = bi;
}

// ------------------------------------------------------------------
// Kernel 3: factored pos GEMM via V_WMMA_F32_16X16X4_F32
// table[r][d] = sum_{r'} (L - |r'-K|) * exp(-(r-r')^2) * W_pos[r'][d]
// M=25 -> 2 tiles of 16, N=50 -> 4 tiles of 16, K=25 -> 7 steps of 4
// ------------------------------------------------------------------
__device__ __forceinline__ float a_elem(int m, int k)
{
    if (m >= (2 * K_ + 1) || k >= (2 * K_ + 1)) return 0.0f;
    const float w = (float)(L_ - ((k > K_) ? (k - K_) : (K_ - k)));
    const int d = m - k;
    return w * expf(-(float)(d * d));
}

__global__ __launch_bounds__(256) void pos_table_kernel(const float* __restrict__ Wpos,
                                                        float* __restrict__ table)
{
    const int lane  = threadIdx.x & 31;
    const int wave  = threadIdx.x >> 5;   // 0..7
    const int mTile = wave >> 2;          // 0..1
    const int nTile = wave & 3;           // 0..3
    const int n     = nTile * 16 + (lane & 15);
    const int m     = mTile * 16 + (lane & 15);
    const int kHalf = (lane >> 4) << 1;   // lanes 0-15 -> 0, lanes 16-31 -> 2

    v8f acc = {};
    for (int kk = 0; kk < 7; ++kk) {
        const int k0 = kk * 4 + kHalf;
        // A fragment: 16x4 f32, lane holds A[m][k0], A[m][k0+1]
        v2f a;
        a.x = a_elem(m, k0);
        a.y = a_elem(m, k0 + 1);
        // B fragment: 4x16 f32, VGPR0 -> row k0, VGPR1 -> row k0+1, col n
        v2f b;
        b.x = (k0     < (2 * K_ + 1) && n < PD) ? Wpos[k0 * PD + n]       : 0.0f;
        b.y = (k0 + 1 < (2 * K_ + 1) && n < PD) ? Wpos[(k0 + 1) * PD + n] : 0.0f;
        acc = __builtin_amdgcn_wmma_f32_16x16x4_f32(false, a, false, b,
                                                    (short)0, acc, false, false);
    }
    // C/D layout: VGPR v, lanes 0-15 -> M=v, lanes 16-31 -> M=v+8
    const int rowBase = mTile * 16 + ((lane >> 4) << 3);
    const int col     = nTile * 16 + (lane & 15);
    #pragma unroll
    for (int v = 0; v < 8; ++v) {
        table[(rowBase + v) * TPAD + col] = acc[v];
    }
}

// ------------------------------------------------------------------
// Kernel 4: main bandwidth kernel — gather rows, concat, NT-stream out
// grid = (NC, B), 256 threads per block; one block per (pair, batch)
// ------------------------------------------------------------------
__global__ __launch_bounds__(256) void pair_copy_kernel(
    const float* __restrict__ he, const float* __restrict__ hc,
    const float* __restrict__ Wemo, const int* __restrict__ ei,
    const int* __restrict__ ci, const int* __restrict__ labels,
    const float* __restrict__ table, float* __restrict__ out)
{
    const int p = blockIdx.x;
    const int b = blockIdx.y;
    const int t = threadIdx.x;

    const int i = ei[p];
    const int j = ci[p];

    const f4* heRow = (const f4*)(he + ((size_t)b * L_ + i) * F_);
    const f4* hcRow = (const f4*)(hc + ((size_t)b * L_ + j) * F_);
    float* dst = out + ((size_t)b * NC + p) * ROW;
    f4* dst4 = (f4*)dst;

    // he segment: float4 [0,192), hc segment: float4 [192,384)
    #pragma unroll
    for (int u = 0; u < 2; ++u) {
        const int idx = t + u * 256;
        if (idx < 384) {
            f4 v = (idx < 192) ? heRow[idx] : hcRow[idx - 192];
            __builtin_nontemporal_store(v, dst4 + idx);
        }
    }
    // emo embedding segment [1536, 1586)
    if (t < ED) {
        const int lab = labels[b * L_ + i];
        __builtin_nontemporal_store(Wemo[lab * ED + t], dst + 1536 + t);
    }
    // pos segment [1586, 1636)
    else if (t >= 64 && t < 64 + PD) {
        const int rel = j - i + K_;
        __builtin_nontemporal_store(table[rel * TPAD + (t - 64)],
                                    dst + 1536 + ED + (t - 64));
    }
}

// ------------------------------------------------------------------
// Host launcher
// ------------------------------------------------------------------
extern "C" void kernel_launch(void* const* d_in, const int* in_sizes, int n_in,
                              void* d_out, int out_size, void* d_ws, size_t ws_size,
                              hipStream_t stream)
{
    const float* he   = (const float*)d_in[0];  // (B, L, F)
    const float* hc   = (const float*)d_in[1];  // (B, L, F)
    const float* pred = (const float*)d_in[2];  // (B, L, TAGSET)
    const float* Wpos = (const float*)d_in[3];  // (2K+1, PD)
    const float* Wemo = (const float*)d_in[4];  // (TAGSET, ED)
    float* out = (float*)d_out;

    char* ws = (char*)d_ws;
    int*   ei     = (int*)(ws + 0);         // NC ints
    int*   ci     = (int*)(ws + 12288);     // NC ints
    int*   labels = (int*)(ws + 24576);     // B*L ints
    float* table  = (float*)(ws + 32768);   // 32*64 floats (padded pos table)

    // emo_cau_pos lives after the big tensor in d_out
    float* out_pos = out + (size_t)B_ * NC * ROW;

    build_pairs_kernel<<<1, 128, 0, stream>>>(ei, ci, out_pos);
    labels_kernel<<<(B_ * L_ + 255) / 256, 256, 0, stream>>>(pred, labels, B_ * L_);
    pos_table_kernel<<<1, 256, 0, stream>>>(Wpos, table);

    dim3 grid(NC, B_);
    pair_copy_kernel<<<grid, 256, 0, stream>>>(he, hc, Wemo, ei, ci, labels,
                                               table, out);
}